// Core_23459111370998
// MI455X (gfx1250) — compile-verified
//
#include <hip/hip_runtime.h>
#include <hip/hip_bf16.h>
#include <math.h>

// ---------------------------------------------------------------------------
// LayerNorm-LSTM for MI455X (gfx1250), fp32 via V_WMMA_F32_16X16X4_F32,
// tile staging via Tensor Data Mover (tensor_load_to_lds + s_wait_tensorcnt).
// T=256, B=64, I=H=512, L=2, G=4H=2048.
// ---------------------------------------------------------------------------

typedef __attribute__((ext_vector_type(2))) float v2f;
typedef __attribute__((ext_vector_type(8))) float v8f;
typedef __attribute__((ext_vector_type(4))) unsigned int u32x4;
typedef __attribute__((ext_vector_type(4))) int i32x4;
typedef __attribute__((ext_vector_type(8))) int i32x8;

#define LSTM_T 256
#define LSTM_B 64
#define LSTM_H 512
#define LSTM_G 2048   // 4*H
#define LSTM_L 2
#define LN_EPS 1e-5f

__device__ __forceinline__ float sigmoidf_(float x) {
  return 1.0f / (1.0f + __expf(-x));
}

// ---------------------------------------------------------------------------
// TDM: 1-D contiguous fp32 copy global -> LDS via tensor descriptor (D#).
// nelem must fit tile_dim0 (16 bits). Issue from ONE wave only; caller waits
// with s_wait_tensorcnt before the workgroup barrier.
// D# layout per CDNA5 ISA.
// ---------------------------------------------------------------------------
__device__ __forceinline__ void tdm_load_f32_1d(void* lds_dst, const void* gsrc,
                                                unsigned nelem) {
  const unsigned lds = (unsigned)(unsigned long long)lds_dst;   // addr[31:0] = LDS offset
  const unsigned long long ga = (unsigned long long)gsrc;

  u32x4 g0;
  g0[0] = 1u;                                        // count=1, user descriptor
  g0[1] = lds;                                       // lds_addr [63:32]
  g0[2] = (unsigned)(ga & 0xffffffffu);              // global_addr [95:64]
  g0[3] = (unsigned)((ga >> 32) & 0x01ffffffu)       // global_addr [120:96]
          | (2u << 30);                              // type=2 ("image")

  i32x8 g1;
  g1[0] = (int)(2u << 16);                           // data_size=2 -> 4 bytes
  g1[1] = (int)((nelem & 0xffffu) << 16);            // tensor_dim0[15:0]
  g1[2] = (int)(((nelem >> 16) & 0xffffu)            // tensor_dim0[31:16]
          | (1u << 16));                             // tensor_dim1 = 1
  g1[3] = (int)((nelem & 0xffffu) << 16);            // tile_dim0 = nelem
  g1[4] = 0;                                         // tile_dim1/2 unused
  g1[5] = (int)nelem;                                // tensor_dim0_stride lo
  g1[6] = 0;
  g1[7] = 0;

  i32x4 z4 = {0, 0, 0, 0};
#if defined(__clang_major__) && __clang_major__ >= 23
  i32x8 z8 = {0, 0, 0, 0, 0, 0, 0, 0};
  __builtin_amdgcn_tensor_load_to_lds(g0, g1, z4, z4, z8, 0);
#else
  __builtin_amdgcn_tensor_load_to_lds(g0, g1, z4, z4, 0);
#endif
}

// ---------------------------------------------------------------------------
// Bulk GEMM: Y[M,N] = X[M,K] @ W[N,K]^T      (K=512, N=2048)
// grid = (M/16, N/128), block = 256 (8 waves); X tile staged in LDS via TDM,
// each wave owns one 16x16 output tile, K-loop of 128 WMMA f32 16x16x4
// (chunked 16x-unrolled with one weight prefetch per 64-element chunk).
// ---------------------------------------------------------------------------
__global__ __launch_bounds__(256)
void gemm_xwt_kernel(const float* __restrict__ X, const float* __restrict__ W,
                     float* __restrict__ Y, int M, int N, int K) {
  __shared__ float sA[16 * LSTM_H];  // 32 KB X tile
  const int tid  = threadIdx.x;
  const int lane = tid & 31;
  const int wave = tid >> 5;           // 0..7
  const int mt   = blockIdx.x;         // 16-row tile index
  const int nb   = blockIdx.y;         // 128-col group

  const float* Xt = X + (size_t)mt * 16 * K;
  if (wave == 0) {
    tdm_load_f32_1d(sA, Xt, 16u * LSTM_H);          // tensor_load_to_lds
    __builtin_amdgcn_s_wait_tensorcnt(0);
  }
  __syncthreads();

  const int ncol0 = nb * 128 + wave * 16;
  const int arow  = lane & 15;           // A: M = lane&15
  const int kb    = (lane >> 4) * 2;     // A/B: K sub-offset 0 or 2
  const int wrow  = ncol0 + (lane & 15); // B: W row (output column)
  const float* Wp = W + (size_t)wrow * K;

  v8f acc = {};
  for (int k0 = 0; k0 < K; k0 += 64) {
    __builtin_prefetch(Wp + k0 + 64, 0, 1);         // global_prefetch_b8
#pragma unroll
    for (int kk = 0; kk < 64; kk += 4) {
      const int k = k0 + kk;
      v2f a = *(const v2f*)(&sA[arow * K + k + kb]);  // ds_load_b64
      v2f b = *(const v2f*)(&Wp[k + kb]);             // global_load_b64
      acc = __builtin_amdgcn_wmma_f32_16x16x4_f32(
          false, a, false, b, (short)0, acc, false, false);
    }
  }

  // C/D layout: VGPR r -> (M=r, lanes0-15) / (M=r+8, lanes16-31), N = lane&15
  const int crow0 = (lane >> 4) * 8;
  const int ccol  = ncol0 + (lane & 15);
  float* Yp = Y + (size_t)(mt * 16 + crow0) * N + ccol;
#pragma unroll
  for (int r = 0; r < 8; ++r) Yp[(size_t)r * N] = acc[r];
}

// ---------------------------------------------------------------------------
// Row LayerNorm in place: Y[row,:] = (y - mean)/sqrt(var+eps)*g + b
// grid = M, block = 256
// ---------------------------------------------------------------------------
__global__ __launch_bounds__(256)
void ln_rows_kernel(float* __restrict__ Y, const float* __restrict__ g,
                    const float* __restrict__ b, int N) {
  __shared__ float red[256];
  __shared__ float red2[256];
  __shared__ float s_mean, s_rstd;
  const int tid = threadIdx.x;
  float* y = Y + (size_t)blockIdx.x * N;

  float s = 0.f, ss = 0.f;
  for (int j = tid; j < N; j += 256) { float v = y[j]; s += v; ss += v * v; }
  red[tid] = s; red2[tid] = ss;
  __syncthreads();
  for (int o = 128; o > 0; o >>= 1) {
    if (tid < o) { red[tid] += red[tid + o]; red2[tid] += red2[tid + o]; }
    __syncthreads();
  }
  if (tid == 0) {
    float mean = red[0] / (float)N;
    float var  = red2[0] / (float)N - mean * mean;
    s_mean = mean;
    s_rstd = rsqrtf(var + LN_EPS);
  }
  __syncthreads();
  const float mean = s_mean, rstd = s_rstd;
  for (int j = tid; j < N; j += 256)
    y[j] = (y[j] - mean) * rstd * g[j] + b[j];
}

// ---------------------------------------------------------------------------
// One LSTM timestep for one layer.  grid = B/16 = 4, block = 256 (8 waves).
// Dynamic LDS: gates[16*2048] | buf[16*512] | red[256] | red2[256] | m/r[32]
// Phases: (1) h-tile -> LDS via TDM, (2) hgates = h @ Whh^T via WMMA -> LDS,
// (3) LN(4H) + add precomputed LN(igates), (4) gate math + c-pre,
// (5) LN(H) on c, (6) c/h update + stores.
// ---------------------------------------------------------------------------
__global__ __launch_bounds__(256)
void lstm_step_kernel(const float* __restrict__ ln_ig_t,  // [B, 4H], LN applied
                      const float* __restrict__ Whh,      // [4H, H]
                      const float* __restrict__ g_h, const float* __restrict__ b_h,
                      const float* __restrict__ g_c, const float* __restrict__ b_c,
                      float* __restrict__ h_state,        // [B, H] in/out
                      float* __restrict__ c_state,        // [B, H] in/out
                      float* __restrict__ out_t)          // [B, H] h output at t
{
  const int H = LSTM_H, G = LSTM_G;
  extern __shared__ float smem[];
  float* gates = smem;                 // 16*2048
  float* buf   = gates + 16 * G;       // 16*512 : h tile, then c-pre
  float* redp  = buf + 16 * H;         // 16*16
  float* red2p = redp + 256;           // 16*16
  float* s_m   = red2p + 256;          // 16
  float* s_r   = s_m + 16;             // 16

  const int tid  = threadIdx.x;
  const int lane = tid & 31;
  const int wave = tid >> 5;
  const int b0   = blockIdx.x * 16;    // batch-row base

  // (1) stage h tile [16, H] via Tensor Data Mover (wave 0 issues, waits)
  if (wave == 0) {
    tdm_load_f32_1d(buf, h_state + (size_t)b0 * H, 16u * LSTM_H);
    __builtin_amdgcn_s_wait_tensorcnt(0);
  }
  __syncthreads();

  // (2) hgates GEMM: 8 waves x 16 n-tiles each (128 tiles of 16 cols)
  const int arow = lane & 15;
  const int kb   = (lane >> 4) * 2;
  for (int ti = 0; ti < 16; ++ti) {
    const int nt   = wave * 16 + ti;
    const int wrow = nt * 16 + (lane & 15);
    const float* Wp = Whh + (size_t)wrow * H;
    v8f acc = {};
    for (int k0 = 0; k0 < H; k0 += 64) {
      __builtin_prefetch(Wp + k0 + 64, 0, 1);
#pragma unroll
      for (int kk = 0; kk < 64; kk += 4) {
        const int k = k0 + kk;
        v2f a  = *(const v2f*)(&buf[arow * H + k + kb]);
        v2f bb = *(const v2f*)(&Wp[k + kb]);
        acc = __builtin_amdgcn_wmma_f32_16x16x4_f32(
            false, a, false, bb, (short)0, acc, false, false);
      }
    }
    const int crow0 = (lane >> 4) * 8;
    const int ccol  = nt * 16 + (lane & 15);
#pragma unroll
    for (int r = 0; r < 8; ++r) gates[(crow0 + r) * G + ccol] = acc[r];
  }
  __syncthreads();

  // (3) LN over hgates rows (4H), then add LN'd input gates
  {
    const int row = tid >> 4, sub = tid & 15;
    float s = 0.f, ss = 0.f;
    for (int j = sub; j < G; j += 16) {
      float v = gates[row * G + j]; s += v; ss += v * v;
    }
    redp[row * 16 + sub] = s; red2p[row * 16 + sub] = ss;
    __syncthreads();
    if (tid < 16) {
      float sm = 0.f, sq = 0.f;
      for (int i = 0; i < 16; ++i) { sm += redp[tid * 16 + i]; sq += red2p[tid * 16 + i]; }
      float mean = sm / (float)G;
      float var  = sq / (float)G - mean * mean;
      s_m[tid] = mean; s_r[tid] = rsqrtf(var + LN_EPS);
    }
    __syncthreads();
    for (int idx = tid; idx < 16 * G; idx += 256) {
      const int row2 = idx >> 11;        // /G
      const int j    = idx & (G - 1);    // %G
      float v = (gates[idx] - s_m[row2]) * s_r[row2] * g_h[j] + b_h[j];
      gates[idx] = v + ln_ig_t[(size_t)(b0 + row2) * G + j];
    }
  }
  __syncthreads();

  // (4) c-pre = sigmoid(f)*c_old + sigmoid(i)*tanh(g)   (buf reused)
  for (int idx = tid; idx < 16 * H; idx += 256) {
    const int row = idx >> 9;          // /H
    const int u   = idx & (H - 1);     // %H
    float ig = gates[row * G + u];
    float fg = gates[row * G + H + u];
    float gg = gates[row * G + 2 * H + u];
    float c_old = c_state[(size_t)(b0 + row) * H + u];
    buf[idx] = sigmoidf_(fg) * c_old + sigmoidf_(ig) * tanhf(gg);
  }
  __syncthreads();

  // (5) LN over c-pre rows (H)
  {
    const int row = tid >> 4, sub = tid & 15;
    float s = 0.f, ss = 0.f;
    for (int j = sub; j < H; j += 16) {
      float v = buf[row * H + j]; s += v; ss += v * v;
    }
    redp[row * 16 + sub] = s; red2p[row * 16 + sub] = ss;
    __syncthreads();
    if (tid < 16) {
      float sm = 0.f, sq = 0.f;
      for (int i = 0; i < 16; ++i) { sm += redp[tid * 16 + i]; sq += red2p[tid * 16 + i]; }
      float mean = sm / (float)H;
      float var  = sq / (float)H - mean * mean;
      s_m[tid] = mean; s_r[tid] = rsqrtf(var + LN_EPS);
    }
    __syncthreads();
  }

  // (6) c_new, h_new = sigmoid(o)*tanh(c_new); write states + sequence out
  for (int idx = tid; idx < 16 * H; idx += 256) {
    const int row = idx >> 9;
    const int u   = idx & (H - 1);
    float cn = (buf[idx] - s_m[row]) * s_r[row] * g_c[u] + b_c[u];
    float og = gates[row * G + 3 * H + u];
    float hn = sigmoidf_(og) * tanhf(cn);
    const size_t gidx = (size_t)(b0 + row) * H + u;
    c_state[gidx] = cn;
    h_state[gidx] = hn;
    out_t[gidx]   = hn;
  }
}

__global__ void copy_f32(float* __restrict__ dst, const float* __restrict__ src, int n) {
  int i = blockIdx.x * 256 + threadIdx.x;
  if (i < n) dst[i] = src[i];
}

// ---------------------------------------------------------------------------
// Host orchestration (all on `stream`, graph-capture safe).
// Inputs: x, h0, c0, w_ih, w_hh, g_i, b_i, g_h, b_h, g_c, b_c
// Output: out[T,B,H] ++ h_last[L,B,H] ++ c_last[L,B,H]
// ---------------------------------------------------------------------------
extern "C" void kernel_launch(void* const* d_in, const int* in_sizes, int n_in,
                              void* d_out, int out_size, void* d_ws, size_t ws_size,
                              hipStream_t stream) {
  (void)in_sizes; (void)n_in; (void)out_size; (void)ws_size;
  const int T = LSTM_T, B = LSTM_B, H = LSTM_H, G = LSTM_G, L = LSTM_L;
  const int I = LSTM_H;  // input dim == H

  const float* x    = (const float*)d_in[0];
  const float* h0   = (const float*)d_in[1];
  const float* c0   = (const float*)d_in[2];
  const float* w_ih = (const float*)d_in[3];
  const float* w_hh = (const float*)d_in[4];
  const float* g_i  = (const float*)d_in[5];
  const float* b_i  = (const float*)d_in[6];
  const float* g_h  = (const float*)d_in[7];
  const float* b_h  = (const float*)d_in[8];
  const float* g_c  = (const float*)d_in[9];
  const float* b_c  = (const float*)d_in[10];

  // workspace carve-up
  float* ws   = (float*)d_ws;
  float* ig   = ws;                             // [T*B, G]   128 MB
  float* seq0 = ig   + (size_t)T * B * G;       // [T*B, H]    32 MB (layer-0 out)
  float* h_st = seq0 + (size_t)T * B * H;       // [B, H]
  float* c_st = h_st + (size_t)B * H;           // [B, H]

  float* out   = (float*)d_out;                 // [T, B, H]
  float* hlast = out   + (size_t)T * B * H;     // [L, B, H]
  float* clast = hlast + (size_t)L * B * H;     // [L, B, H]

  const size_t step_lds =
      (size_t)(16 * G + 16 * H + 256 + 256 + 32) * sizeof(float);

  for (int l = 0; l < L; ++l) {
    const float* Xl  = (l == 0) ? x : seq0;
    const float* Wih = w_ih + (size_t)l * G * I;
    const float* Whh = w_hh + (size_t)l * G * H;
    float* layer_out = (l == L - 1) ? out : seq0;

    // bulk input-gate GEMM for all timesteps, then row LayerNorm
    dim3 ggrid(T * B / 16, G / 128);
    gemm_xwt_kernel<<<ggrid, 256, 0, stream>>>(Xl, Wih, ig, T * B, G, I);
    ln_rows_kernel<<<T * B, 256, 0, stream>>>(ig, g_i + (size_t)l * G,
                                              b_i + (size_t)l * G, G);

    // init recurrent state
    copy_f32<<<(B * H + 255) / 256, 256, 0, stream>>>(h_st, h0 + (size_t)l * B * H, B * H);
    copy_f32<<<(B * H + 255) / 256, 256, 0, stream>>>(c_st, c0 + (size_t)l * B * H, B * H);

    // sequential recurrence
    for (int t = 0; t < T; ++t) {
      lstm_step_kernel<<<B / 16, 256, step_lds, stream>>>(
          ig + (size_t)t * B * G, Whh,
          g_h + (size_t)l * G, b_h + (size_t)l * G,
          g_c + (size_t)l * H, b_c + (size_t)l * H,
          h_st, c_st, layer_out + (size_t)t * B * H);
    }

    // final states
    copy_f32<<<(B * H + 255) / 256, 256, 0, stream>>>(hlast + (size_t)l * B * H, h_st, B * H);
    copy_f32<<<(B * H + 255) / 256, 256, 0, stream>>>(clast + (size_t)l * B * H, c_st, B * H);
  }
}